// GraphSAGEModel_60790967107705
// MI455X (gfx1250) — compile-verified
//
#include <hip/hip_runtime.h>
#include <hip/hip_bf16.h>

#define N_NODES   100000
#define N_EDGES   600000
#define DIM       128
#define N_LAYERS  3
#define N_CLASSES 2
#define BN_EPS    1e-5f

typedef float v2f __attribute__((ext_vector_type(2)));
typedef float v8f __attribute__((ext_vector_type(8)));

// ---------------------------------------------------------------------------
// Degree count:  cnt[dst[e]] += 1   (cnt zeroed by memset before)
// ---------------------------------------------------------------------------
__global__ void count_deg_kernel(const long long* __restrict__ dst,
                                 float* __restrict__ cnt) {
    int e = blockIdx.x * blockDim.x + threadIdx.x;
    if (e < N_EDGES) atomicAdd(&cnt[(int)dst[e]], 1.0f);
}

// In-place cnt -> inv_cnt
__global__ void inv_deg_kernel(float* __restrict__ cnt) {
    int i = blockIdx.x * blockDim.x + threadIdx.x;
    if (i < N_NODES) {
        float c = cnt[i];
        cnt[i] = (c > 0.0f) ? (1.0f / c) : 0.0f;
    }
}

// ---------------------------------------------------------------------------
// Scatter aggregation: agg[dst[e], :] += x[src[e], :]
// 32 threads per edge, 4 consecutive features per thread (float4 gather).
// ---------------------------------------------------------------------------
__global__ void scatter_agg_kernel(const long long* __restrict__ src,
                                   const long long* __restrict__ dst,
                                   const float* __restrict__ x,
                                   float* __restrict__ agg) {
    int tid = blockIdx.x * blockDim.x + threadIdx.x;
    int e   = tid >> 5;
    if (e >= N_EDGES) return;
    int f   = (tid & 31) << 2;           // feature offset 0..124
    int s   = (int)src[e];
    int d   = (int)dst[e];
    float4 v = *(const float4*)&x[(long long)s * DIM + f];
    float* base = &agg[(long long)d * DIM + f];
    atomicAdd(base + 0, v.x);
    atomicAdd(base + 1, v.y);
    atomicAdd(base + 2, v.z);
    atomicAdd(base + 3, v.w);
}

// ---------------------------------------------------------------------------
// Fused SAGE layer:
//   x_out = relu( BN( (agg*inv)@Wl^T + bl + x@Wr^T ) ) + residual
//   residual = x@proj^T on layer 0, x otherwise.
//
// Block = 256 threads = 8 waves. Block owns 16 rows (nodes); wave w owns
// column tile [16w, 16w+16). A-tiles (agg*inv, x) staged in LDS with a
// 132-float padded stride (conflict free, 16B aligned). K walked in steps
// of 4 with V_WMMA_F32_16X16X4_F32.
//
// WMMA f32 16x16x4 operand layout (ISA 7.12.2):
//   A (16x4): lane m = lane&15, v0 = A[m][2*half+0], v1 = A[m][2*half+1]
//   B (4x16): lane n = lane&15, v0 = B[2*half+0][n], v1 = B[2*half+1][n]
//   C/D     : vgpr r, lanes 0-15 -> M=r, lanes 16-31 -> M=r+8, N=lane&15
// ---------------------------------------------------------------------------
#define LDS_STRIDE 132

template <int LAYER0>
__global__ __launch_bounds__(256)
void sage_layer_kernel(const float* __restrict__ x_in,
                       const float* __restrict__ agg,
                       const float* __restrict__ invc,
                       const float* __restrict__ Wl,    // [DIM,DIM]  (n,k)
                       const float* __restrict__ bl,    // [DIM]
                       const float* __restrict__ Wr,    // [DIM,DIM]
                       const float* __restrict__ bng,
                       const float* __restrict__ bnb,
                       const float* __restrict__ bnm,
                       const float* __restrict__ bnv,
                       const float* __restrict__ projW, // [DIM,DIM]
                       float* __restrict__ x_out) {
    __shared__ float sAgg[16 * LDS_STRIDE];
    __shared__ float sX[16 * LDS_STRIDE];

    const int row0 = blockIdx.x * 16;

    // Cooperative stage of the 16xDIM A-tiles (512 float4 per matrix).
    for (int j = threadIdx.x; j < 512; j += 256) {
        int r  = j >> 5;            // row in tile
        int kq = (j & 31) << 2;     // k offset, step 4
        long long g = (long long)(row0 + r) * DIM + kq;
        float4 a = *(const float4*)&agg[g];
        float ic = invc[row0 + r];
        a.x *= ic; a.y *= ic; a.z *= ic; a.w *= ic;
        *(float4*)&sAgg[r * LDS_STRIDE + kq] = a;
        *(float4*)&sX[r * LDS_STRIDE + kq]   = *(const float4*)&x_in[g];
    }
    __syncthreads();

    const int wave  = threadIdx.x >> 5;
    const int lane  = threadIdx.x & 31;
    const int col0  = wave << 4;
    const int m     = lane & 15;
    const int khalf = lane >> 4;
    const int n     = col0 + m;     // output column this lane owns

    v8f acc  = {};
    v8f acc2 = {};

#pragma unroll
    for (int kk = 0; kk < DIM; kk += 4) {
        int kb = kk + khalf * 2;
        v2f aAgg = *(const v2f*)&sAgg[m * LDS_STRIDE + kb];
        v2f aX   = *(const v2f*)&sX[m * LDS_STRIDE + kb];
        v2f bWl  = *(const v2f*)&Wl[n * DIM + kb];
        v2f bWr  = *(const v2f*)&Wr[n * DIM + kb];
        acc = __builtin_amdgcn_wmma_f32_16x16x4_f32(false, aAgg, false, bWl,
                                                    (short)0, acc, false, false);
        acc = __builtin_amdgcn_wmma_f32_16x16x4_f32(false, aX, false, bWr,
                                                    (short)0, acc, false, false);
        if (LAYER0) {
            v2f bPj = *(const v2f*)&projW[n * DIM + kb];
            acc2 = __builtin_amdgcn_wmma_f32_16x16x4_f32(false, aX, false, bPj,
                                                         (short)0, acc2, false, false);
        }
    }

    // BN (inference) folded to per-column scale/shift; bl folded in too.
    float scale = bng[n] * rsqrtf(bnv[n] + BN_EPS);
    float shift = (bl[n] - bnm[n]) * scale + bnb[n];

#pragma unroll
    for (int r = 0; r < 8; ++r) {
        int mloc = r + khalf * 8;               // row within tile
        int mm   = row0 + mloc;
        float h  = fmaxf(acc[r] * scale + shift, 0.0f);
        float res = LAYER0 ? acc2[r] : sX[mloc * LDS_STRIDE + n];
        x_out[(long long)mm * DIM + n] = h + res;
    }
}

// ---------------------------------------------------------------------------
// Classifier: out[i,c] = x[i,:] . cls_W[c,:] + cls_b[c]    (C = 2)
// ---------------------------------------------------------------------------
__global__ void classifier_kernel(const float* __restrict__ x,
                                  const float* __restrict__ W,
                                  const float* __restrict__ b,
                                  float* __restrict__ out) {
    int i = blockIdx.x * blockDim.x + threadIdx.x;
    if (i >= N_NODES) return;
    const float* xr = &x[(long long)i * DIM];
    float a0 = 0.0f, a1 = 0.0f;
#pragma unroll 8
    for (int k = 0; k < DIM; k += 4) {
        float4 v  = *(const float4*)&xr[k];
        float4 w0 = *(const float4*)&W[k];
        float4 w1 = *(const float4*)&W[DIM + k];
        a0 += v.x * w0.x + v.y * w0.y + v.z * w0.z + v.w * w0.w;
        a1 += v.x * w1.x + v.y * w1.y + v.z * w1.z + v.w * w1.w;
    }
    out[(long long)i * N_CLASSES + 0] = a0 + b[0];
    out[(long long)i * N_CLASSES + 1] = a1 + b[1];
}

// ---------------------------------------------------------------------------
extern "C" void kernel_launch(void* const* d_in, const int* in_sizes, int n_in,
                              void* d_out, int out_size, void* d_ws, size_t ws_size,
                              hipStream_t stream) {
    const float*     x    = (const float*)d_in[0];
    const long long* ei   = (const long long*)d_in[1];   // int64 [2, E]
    const long long* src  = ei;
    const long long* dst  = ei + N_EDGES;
    const float*     Wl   = (const float*)d_in[2];       // [L, D, D]
    const float*     bl   = (const float*)d_in[3];       // [L, D]
    const float*     Wr   = (const float*)d_in[4];       // [L, D, D]
    const float*     bng  = (const float*)d_in[5];
    const float*     bnb  = (const float*)d_in[6];
    const float*     bnm  = (const float*)d_in[7];
    const float*     bnv  = (const float*)d_in[8];
    const float*     prj  = (const float*)d_in[9];       // [D, D]
    const float*     clsW = (const float*)d_in[10];      // [2, D]
    const float*     clsb = (const float*)d_in[11];      // [2]
    float*           out  = (float*)d_out;

    // Workspace layout (256B-aligned blocks)
    const size_t XBYTES = (size_t)N_NODES * DIM * sizeof(float);   // 51.2 MB
    char* ws = (char*)d_ws;
    float* invc = (float*)ws;                                      // N floats
    float* agg  = (float*)(ws + (1 << 19));                        // +512 KB
    float* xA   = (float*)(ws + (1 << 19) + XBYTES);
    float* xB   = (float*)(ws + (1 << 19) + 2 * XBYTES);

    // Degree counts -> inverse
    hipMemsetAsync(invc, 0, (size_t)N_NODES * sizeof(float), stream);
    count_deg_kernel<<<(N_EDGES + 255) / 256, 256, 0, stream>>>(dst, invc);
    inv_deg_kernel<<<(N_NODES + 255) / 256, 256, 0, stream>>>(invc);

    const float* cur = x;
    float* nxt = xA;
    for (int i = 0; i < N_LAYERS; ++i) {
        hipMemsetAsync(agg, 0, XBYTES, stream);
        int sc_threads = N_EDGES * 32;
        scatter_agg_kernel<<<(sc_threads + 255) / 256, 256, 0, stream>>>(src, dst, cur, agg);

        dim3 grid(N_NODES / 16);
        const float* Wli = Wl + (size_t)i * DIM * DIM;
        const float* Wri = Wr + (size_t)i * DIM * DIM;
        if (i == 0) {
            sage_layer_kernel<1><<<grid, 256, 0, stream>>>(
                cur, agg, invc, Wli, bl + i * DIM, Wri,
                bng + i * DIM, bnb + i * DIM, bnm + i * DIM, bnv + i * DIM,
                prj, nxt);
        } else {
            sage_layer_kernel<0><<<grid, 256, 0, stream>>>(
                cur, agg, invc, Wli, bl + i * DIM, Wri,
                bng + i * DIM, bnb + i * DIM, bnm + i * DIM, bnv + i * DIM,
                prj, nxt);
        }
        cur = nxt;
        nxt = (nxt == xA) ? xB : xA;
    }

    classifier_kernel<<<(N_NODES + 255) / 256, 256, 0, stream>>>(cur, clsW, clsb, out);
}